// ume_kp_layer_6597069766923
// MI455X (gfx1250) — compile-verified
//
#include <hip/hip_runtime.h>
#include <hip/hip_bf16.h>

#define NKP   64
#define KNN   64
#define FDIM  64
#define RAD2  (1.5f * 1.5f)
#define EPS_M 1e-6f
#define EPS_T 1e-16f

typedef __attribute__((ext_vector_type(2))) float v2f;
typedef __attribute__((ext_vector_type(8))) float v8f;

// ---------------------------------------------------------------------------
// Stage 1: ball query (first-64 in-radius by index order, == sort semantics of
// the reference) + UME moment matrix G/H (F x 4, normalized).
// One 256-thread block per (kp, batch, src/tgt). Wave 0 builds neighbor list
// with wave32 ballot prefix scan; all 256 threads then compute um[f][d].
// ---------------------------------------------------------------------------
__global__ void k_ball_ume(const float* __restrict__ s_pts, const float* __restrict__ s_ft,
                           const float* __restrict__ s_kp,  const float* __restrict__ t_pts,
                           const float* __restrict__ t_ft,  const float* __restrict__ t_kp,
                           float* __restrict__ outG, float* __restrict__ outH,
                           float* __restrict__ Gcol, float* __restrict__ Hrow, int N) {
  const int kpi = blockIdx.x, b = blockIdx.y, which = blockIdx.z;
  const float* pts = which ? t_pts : s_pts;
  const float* ft  = which ? t_ft  : s_ft;
  const float* kp  = which ? t_kp  : s_kp;

  __shared__ int   nbr[KNN];
  __shared__ int   sh_cnt;
  __shared__ float red[FDIM];
  __shared__ float sh_denom;

  const int t = threadIdx.x;
  const float kx = kp[(b * NKP + kpi) * 3 + 0];
  const float ky = kp[(b * NKP + kpi) * 3 + 1];
  const float kz = kp[(b * NKP + kpi) * 3 + 2];

  if (t < 32) {  // wave 0: ordered in-radius index scan
    int cnt = 0;
    const unsigned long long lt = (1ull << t) - 1ull;
    for (int base = 0; base < N && cnt < KNN; base += 32) {
      const int p = base + t;  // N is a multiple of 32
      const float* pp = pts + (size_t)(b * N + p) * 3;
      const float dx = kx - pp[0], dy = ky - pp[1], dz = kz - pp[2];
      const bool in = (dx * dx + dy * dy + dz * dz) < RAD2;
      const unsigned long long m = __ballot(in);
      const int pos = cnt + (int)__popcll(m & lt);
      if (in && pos < KNN) nbr[pos] = p;
      cnt += (int)__popcll(m);
    }
    if (t == 0) sh_cnt = cnt < KNN ? cnt : KNN;
  }
  __syncthreads();

  const int cnt = sh_cnt;
  const int f = t & 63, d = t >> 6;  // um[f][d], d==0 -> feature sum, d>0 -> feat*coord
  float acc = 0.f;
  for (int k = 0; k < cnt; ++k) {
    const int p = nbr[k];
    const float fv = ft[(size_t)(b * N + p) * FDIM + f];
    const float cv = (d == 0) ? 1.f : pts[(size_t)(b * N + p) * 3 + (d - 1)];
    acc += fv * cv;
  }
  if (d == 0) red[f] = acc;
  __syncthreads();
  if (t == 0) {
    float s = 0.f;
    for (int i = 0; i < FDIM; ++i) s += red[i];
    sh_denom = s + EPS_M;
  }
  __syncthreads();
  const float val = acc / sh_denom;

  float* outGH = which ? outH : outG;
  outGH[(((size_t)b * NKP + kpi) * FDIM + f) * 4 + d] = val;  // (bs,nkp,F,4) output
  if (which == 0) Gcol[((size_t)b * FDIM + f) * 256 + 4 * kpi + d] = val;  // B operand: (f, 4i+b)
  else            Hrow[((size_t)b * 256 + 4 * kpi + d) * FDIM + f] = val;  // A operand: (4j+a, f)
}

// ---------------------------------------------------------------------------
// Stage 2: per-kp reduced QR (modified Gram-Schmidt, 64x4) + G-gram col0.
// One 64-thread block per (kp, batch, src/tgt); thread r owns row r.
// ---------------------------------------------------------------------------
__device__ __forceinline__ float blockReduce64(float x, volatile float* s, int t) {
  s[t] = x; __syncthreads();
  if (t < 32) s[t] += s[t + 32]; __syncthreads();
  if (t < 16) s[t] += s[t + 16]; __syncthreads();
  if (t <  8) s[t] += s[t +  8]; __syncthreads();
  if (t <  4) s[t] += s[t +  4]; __syncthreads();
  if (t <  2) s[t] += s[t +  2]; __syncthreads();
  if (t <  1) s[t] += s[t +  1]; __syncthreads();
  const float r = s[0]; __syncthreads();
  return r;
}

__global__ void k_qr(const float* __restrict__ outG, const float* __restrict__ outH,
                     float* __restrict__ QGcol, float* __restrict__ QHrow,
                     float* __restrict__ GG0) {
  const int kpi = blockIdx.x, b = blockIdx.y, which = blockIdx.z;
  const int r = threadIdx.x;  // 0..63
  __shared__ float s[64];
  const float* M = (which ? outH : outG) + ((size_t)(b * NKP + kpi) * FDIM) * 4;
  float v[4];
  for (int c = 0; c < 4; ++c) v[c] = M[r * 4 + c];

  if (which == 0) {  // mg_sq / gmg terms: GG0[c] = <col0, colc>
    for (int c = 0; c < 4; ++c) {
      const float dot = blockReduce64(v[0] * v[c], s, r);
      if (r == 0) GG0[((size_t)b * NKP + kpi) * 4 + c] = dot;
    }
  }
  for (int c = 0; c < 4; ++c) {  // MGS (span == jnp.linalg.qr span; D depends only on span)
    const float nrm = blockReduce64(v[c] * v[c], s, r);
    const float inv = rsqrtf(nrm + 1e-30f);
    v[c] *= inv;
    for (int c2 = c + 1; c2 < 4; ++c2) {
      const float dt = blockReduce64(v[c] * v[c2], s, r);
      v[c2] -= dt * v[c];
    }
  }
  if (which == 0) { for (int c = 0; c < 4; ++c) QGcol[((size_t)b * FDIM + r) * 256 + 4 * kpi + c] = v[c]; }
  else            { for (int c = 0; c < 4; ++c) QHrow[((size_t)b * 256 + 4 * kpi + c) * FDIM + r] = v[c]; }
}

// ---------------------------------------------------------------------------
// Stage 3: the two batched GEMMs, 256x64 @ 64x256 f32, via V_WMMA_F32_16X16X4_F32.
//   which==0: W = Hrow x Gcol  (all pair-wise moment cross-terms)
//   which==1: C = QHrow x QGcol (QH^T QG blocks for the D metric)
// One wave (32 lanes, full EXEC) per 16x16 output tile; 16 K=4 WMMA steps.
// ---------------------------------------------------------------------------
__global__ void k_wmma(const float* __restrict__ Hrow, const float* __restrict__ Gcol,
                       const float* __restrict__ QHrow, const float* __restrict__ QGcol,
                       float* __restrict__ Wbuf, float* __restrict__ Cbuf) {
  const int tile = blockIdx.x, b = blockIdx.y, which = blockIdx.z;
  const float* A = (which ? QHrow : Hrow) + (size_t)b * 256 * 64;
  const float* B = (which ? QGcol : Gcol) + (size_t)b * 64 * 256;
  float*       O = (which ? Cbuf  : Wbuf) + (size_t)b * 256 * 256;

  const int tm = (tile >> 4) << 4, tn = (tile & 15) << 4;
  const int lane = threadIdx.x & 31;
  const int r = lane & 15, h = lane >> 4;

  v8f acc = {};
  const float* Arow = A + (size_t)(tm + r) * 64;
#pragma unroll
  for (int k0 = 0; k0 < 64; k0 += 4) {
    const int ka = k0 + 2 * h;  // ISA 16x4 f32 A layout: K = 2*(lane>>4) + vgpr
    v2f av, bv;
    av.x = Arow[ka];
    av.y = Arow[ka + 1];
    bv.x = B[(size_t)ka * 256 + tn + r];
    bv.y = B[(size_t)(ka + 1) * 256 + tn + r];
    acc = __builtin_amdgcn_wmma_f32_16x16x4_f32(false, av, false, bv, (short)0, acc,
                                                false, false);
  }
#pragma unroll
  for (int vv = 0; vv < 8; ++vv)  // C/D layout: row = vv + 8h, col = lane&15
    O[(size_t)(tm + vv + 8 * h) * 256 + tn + r] = acc[vv];
}

// ---------------------------------------------------------------------------
// Stage 4: per-pair finalize. Read 4x4 W block + per-i gram -> wlc/wrc -> 3x3 M
// -> Jacobi SVD of M^T -> Kabsch R with det-sign fix -> T. D from the C block:
// D = 0.707*sqrt(8 - 2*||QH^T QG||_F^2).
// ---------------------------------------------------------------------------
__global__ void k_fin(const float* __restrict__ Wbuf, const float* __restrict__ Cbuf,
                      const float* __restrict__ GG0, float* __restrict__ outT,
                      float* __restrict__ outD, int bs) {
  const int tid = blockIdx.x * blockDim.x + threadIdx.x;
  const int total = bs * NKP * NKP;
  if (tid >= total) return;
  const int b = tid / (NKP * NKP);
  const int rem = tid - b * NKP * NKP;
  const int i = rem >> 6, j = rem & 63;  // G <- source kp i, H <- target kp j

  const float* Wb = Wbuf + (size_t)b * 256 * 256;
  float W4[4][4];
  for (int a = 0; a < 4; ++a)
    for (int c = 0; c < 4; ++c) W4[a][c] = Wb[(size_t)(4 * j + a) * 256 + 4 * i + c];

  const float* gg = GG0 + ((size_t)b * NKP + i) * 4;
  const float mg_sq = gg[0] + EPS_T;
  float wlc[3], wrc[3];
  for (int x = 0; x < 3; ++x) wlc[x] = gg[x + 1] / (mg_sq + EPS_T);
  const float mg_mh = W4[0][0];
  for (int x = 0; x < 3; ++x) wrc[x] = W4[x + 1][0] / (mg_mh + EPS_T);

  // M[a][c] = sum_f right[f,a]*left[f,c], expanded through the W cross-terms
  float Mm[3][3];
  for (int a = 0; a < 3; ++a)
    for (int c = 0; c < 3; ++c)
      Mm[a][c] = W4[a + 1][c + 1] - wlc[c] * W4[a + 1][0] - wrc[a] * W4[0][c + 1]
               + wrc[a] * wlc[c] * W4[0][0];

  float Am[3][3];  // SVD target = M^T (reference: svd(swapaxes(M,1,2)))
  for (int a = 0; a < 3; ++a)
    for (int c = 0; c < 3; ++c) Am[a][c] = Mm[c][a];

  // Jacobi eigensolve of A^T A -> V, then U = A V / sigma
  float B3[3][3];
  for (int a = 0; a < 3; ++a)
    for (int c = 0; c < 3; ++c) {
      float s2 = 0.f;
      for (int k = 0; k < 3; ++k) s2 += Am[k][a] * Am[k][c];
      B3[a][c] = s2;
    }
  float V[3][3] = {{1, 0, 0}, {0, 1, 0}, {0, 0, 1}};
  for (int sweep = 0; sweep < 6; ++sweep) {
    for (int pi = 0; pi < 3; ++pi) {
      const int p = (pi == 2) ? 1 : 0;
      const int q = (pi == 0) ? 1 : 2;
      const float apq = B3[p][q];
      if (fabsf(apq) > 1e-20f) {
        const float tau = (B3[q][q] - B3[p][p]) / (2.f * apq);
        const float tt = (tau >= 0.f ? 1.f : -1.f) / (fabsf(tau) + sqrtf(1.f + tau * tau));
        const float cc = rsqrtf(1.f + tt * tt);
        const float ss = tt * cc;
        for (int k = 0; k < 3; ++k) {
          const float bkp = B3[k][p], bkq = B3[k][q];
          B3[k][p] = cc * bkp - ss * bkq;
          B3[k][q] = ss * bkp + cc * bkq;
        }
        for (int k = 0; k < 3; ++k) {
          const float bpk = B3[p][k], bqk = B3[q][k];
          B3[p][k] = cc * bpk - ss * bqk;
          B3[q][k] = ss * bpk + cc * bqk;
        }
        for (int k = 0; k < 3; ++k) {
          const float vp = V[k][p], vq = V[k][q];
          V[k][p] = cc * vp - ss * vq;
          V[k][q] = ss * vp + cc * vq;
        }
      }
    }
  }
  float U[3][3], sig[3];
  for (int k = 0; k < 3; ++k) {
    float u0 = 0.f, u1 = 0.f, u2 = 0.f;
    for (int m = 0; m < 3; ++m) {
      u0 += Am[0][m] * V[m][k];
      u1 += Am[1][m] * V[m][k];
      u2 += Am[2][m] * V[m][k];
    }
    const float sg = sqrtf(u0 * u0 + u1 * u1 + u2 * u2);
    sig[k] = sg;
    const float inv = 1.f / fmaxf(sg, 1e-20f);
    U[0][k] = u0 * inv; U[1][k] = u1 * inv; U[2][k] = u2 * inv;
  }
  for (int a = 0; a < 2; ++a)  // sort singular values descending (swap U/V cols)
    for (int c = 0; c < 2 - a; ++c)
      if (sig[c] < sig[c + 1]) {
        const float tmp = sig[c]; sig[c] = sig[c + 1]; sig[c + 1] = tmp;
        for (int k = 0; k < 3; ++k) {
          float tu = U[k][c]; U[k][c] = U[k][c + 1]; U[k][c + 1] = tu;
          float tv = V[k][c]; V[k][c] = V[k][c + 1]; V[k][c + 1] = tv;
        }
      }
  const float detA = Am[0][0] * (Am[1][1] * Am[2][2] - Am[1][2] * Am[2][1])
                   - Am[0][1] * (Am[1][0] * Am[2][2] - Am[1][2] * Am[2][0])
                   + Am[0][2] * (Am[1][0] * Am[2][1] - Am[1][1] * Am[2][0]);
  const float s3 = (detA >= 0.f) ? 1.f : -1.f;  // sign(det(U VH)) == sign(det A)

  float R[3][3];
  for (int r2 = 0; r2 < 3; ++r2)
    for (int c = 0; c < 3; ++c)
      R[r2][c] = U[r2][0] * V[c][0] + U[r2][1] * V[c][1] + s3 * U[r2][2] * V[c][2];

  float b2[3];
  for (int c = 0; c < 3; ++c)
    b2[c] = wrc[c] - (wlc[0] * R[0][c] + wlc[1] * R[1][c] + wlc[2] * R[2][c]);

  float* To = outT + (size_t)tid * 16;  // T[b][i][j][4][4]
  for (int r2 = 0; r2 < 3; ++r2) {
    for (int c = 0; c < 3; ++c) To[r2 * 4 + c] = R[c][r2];  // R^T
    To[r2 * 4 + 3] = b2[r2];
  }
  To[12] = 0.f; To[13] = 0.f; To[14] = 0.f; To[15] = 1.f;

  const float* Cb = Cbuf + (size_t)b * 256 * 256;
  float cs = 0.f;
  for (int a = 0; a < 4; ++a)
    for (int c = 0; c < 4; ++c) {
      const float x = Cb[(size_t)(4 * j + a) * 256 + 4 * i + c];
      cs += x * x;
    }
  outD[tid] = 0.707f * sqrtf(fmaxf(8.f - 2.f * cs, 0.f));
}

// ---------------------------------------------------------------------------
extern "C" void kernel_launch(void* const* d_in, const int* in_sizes, int n_in,
                              void* d_out, int out_size, void* d_ws, size_t ws_size,
                              hipStream_t stream) {
  const float* s_pts = (const float*)d_in[0];
  const float* s_ft  = (const float*)d_in[1];
  const float* s_kp  = (const float*)d_in[2];
  const float* t_pts = (const float*)d_in[3];
  const float* t_ft  = (const float*)d_in[4];
  const float* t_kp  = (const float*)d_in[5];

  const int bs = in_sizes[2] / (NKP * 3);
  const int N  = in_sizes[0] / (bs * 3);

  float* out = (float*)d_out;
  const size_t nT = (size_t)bs * NKP * NKP * 16;
  const size_t nD = (size_t)bs * NKP * NKP;
  const size_t nG = (size_t)bs * NKP * FDIM * 4;
  float* outT = out;
  float* outD = out + nT;
  float* outG = out + nT + nD;
  float* outH = out + nT + nD + nG;

  float* ws = (float*)d_ws;
  size_t o = 0;
  float* Gcol  = ws + o; o += (size_t)bs * FDIM * 256;   // (b, f, 4i+b2)
  float* Hrow  = ws + o; o += (size_t)bs * 256 * FDIM;   // (b, 4j+a, f)
  float* QGcol = ws + o; o += (size_t)bs * FDIM * 256;
  float* QHrow = ws + o; o += (size_t)bs * 256 * FDIM;
  float* Wbuf  = ws + o; o += (size_t)bs * 256 * 256;
  float* Cbuf  = ws + o; o += (size_t)bs * 256 * 256;
  float* GG0   = ws + o; o += (size_t)bs * NKP * 4;

  k_ball_ume<<<dim3(NKP, bs, 2), 256, 0, stream>>>(s_pts, s_ft, s_kp, t_pts, t_ft, t_kp,
                                                   outG, outH, Gcol, Hrow, N);
  k_qr<<<dim3(NKP, bs, 2), 64, 0, stream>>>(outG, outH, QGcol, QHrow, GG0);
  k_wmma<<<dim3(256, bs, 2), 32, 0, stream>>>(Hrow, Gcol, QHrow, QGcol, Wbuf, Cbuf);
  const int tot = bs * NKP * NKP;
  k_fin<<<(tot + 255) / 256, 256, 0, stream>>>(Wbuf, Cbuf, GG0, outT, outD, bs);
}